// Z_pereodic_gen_86801289052188
// MI455X (gfx1250) — compile-verified
//
#include <hip/hip_runtime.h>
#include <math.h>

typedef float v2f __attribute__((ext_vector_type(2)));
typedef float v4f __attribute__((ext_vector_type(4)));
typedef float v8f __attribute__((ext_vector_type(8)));

#define B_  64
#define G_  512
#define H_  512
#define P_  64
#define S_  256
#define INV2PI 0.15915493667125702f   // 1/(2*pi)

// ---------------------------------------------------------------------------
// Kernel A: X = relu(Zg @ Wl + bl)          X: [64,512] f32 (workspace)
// One wave per 16x16 output tile; K stepped by 4 with V_WMMA_F32_16X16X4_F32.
// grid = (64/16)*(512/16) = 128 blocks of 32 threads (1 wave each).
// ---------------------------------------------------------------------------
__global__ __launch_bounds__(32)
void mlp_hidden_kernel(const float* __restrict__ Zg, const float* __restrict__ Wl,
                       const float* __restrict__ bl, float* __restrict__ X) {
  const int tile = blockIdx.x;
  const int mt   = tile & 3;       // M tile: 0..3
  const int nt   = tile >> 2;      // N tile: 0..31
  const int lane = threadIdx.x;    // wave32
  const int mr   = lane & 15;      // A: lane -> M row
  const int kh   = lane >> 4;      // half selects K pair {0,1} vs {2,3}
  const int col  = nt * 16 + mr;   // B/C/D: lane&15 -> N col

  v8f acc = {};
  const float* arow = Zg + (mt * 16 + mr) * G_ + 2 * kh;  // A row-major [64,512]
  const float* bcol = Wl + (2 * kh) * H_ + col;           // B row-major [512,512]
  for (int k0 = 0; k0 < G_; k0 += 4) {
    v2f a = *(const v2f*)(arow + k0);        // A[m][k0+2kh .. +1], one b64 load
    v2f b;
    b.x = bcol[k0 * H_];                     // B[k0+2kh  ][col]
    b.y = bcol[k0 * H_ + H_];                // B[k0+2kh+1][col]
    acc = __builtin_amdgcn_wmma_f32_16x16x4_f32(
        /*neg_a=*/false, a, /*neg_b=*/false, b,
        /*c_mod=*/(short)0, acc, /*reuse_a=*/false, /*reuse_b=*/false);
  }
  const float bias = bl[col];
#pragma unroll
  for (int r = 0; r < 8; ++r) {              // C/D: VGPR r -> M = r + 8*kh
    float v = acc[r] + bias;
    v = v > 0.f ? v : 0.f;                   // relu
    X[(mt * 16 + 8 * kh + r) * H_ + col] = v;
  }
}

// ---------------------------------------------------------------------------
// Kernel B: K1c = (X@W1 + b1)/(2pi), K2c = (X@W2 + b2)/(2pi)   each [64,64]
// grid = 2 heads * 4 mt * 4 nt = 32 blocks of 32 threads.
// ---------------------------------------------------------------------------
__global__ __launch_bounds__(32)
void mlp_heads_kernel(const float* __restrict__ X,
                      const float* __restrict__ W1, const float* __restrict__ b1,
                      const float* __restrict__ W2, const float* __restrict__ b2,
                      float* __restrict__ K1c, float* __restrict__ K2c) {
  const int t    = blockIdx.x;
  const int head = t >> 4;
  const int mt   = (t >> 2) & 3;
  const int nt   = t & 3;
  const float* W  = head ? W2 : W1;
  const float* bb = head ? b2 : b1;
  float*       K  = head ? K2c : K1c;

  const int lane = threadIdx.x;
  const int mr   = lane & 15;
  const int kh   = lane >> 4;
  const int col  = nt * 16 + mr;

  v8f acc = {};
  const float* arow = X + (mt * 16 + mr) * H_ + 2 * kh;   // [64,512]
  const float* bcol = W + (2 * kh) * P_ + col;            // [512,64]
  for (int k0 = 0; k0 < H_; k0 += 4) {
    v2f a = *(const v2f*)(arow + k0);
    v2f b;
    b.x = bcol[k0 * P_];
    b.y = bcol[k0 * P_ + P_];
    acc = __builtin_amdgcn_wmma_f32_16x16x4_f32(
        false, a, false, b, (short)0, acc, false, false);
  }
  const float bias = bb[col];
#pragma unroll
  for (int r = 0; r < 8; ++r) {
    // pre-scale into revolutions for v_sin_f32 (sin of 2*pi*x)
    K[(mt * 16 + 8 * kh + r) * P_ + col] = (acc[r] + bias) * INV2PI;
  }
}

// ---------------------------------------------------------------------------
// Kernel C (bandwidth-dominant): out[b,p,l,m] = sin(2pi*(K1c*l + K2c*m + phi))
// One block per (b,p) pair; 256 threads; thread owns 4 consecutive m values,
// 4 rows per iteration -> 4 KB contiguous NT stores per block-iteration.
// ---------------------------------------------------------------------------
__global__ __launch_bounds__(256)
void periodic_sin_kernel(const float* __restrict__ K1c, const float* __restrict__ K2c,
                         const float* __restrict__ phi, float* __restrict__ out) {
  const int bp  = blockIdx.x;               // b*P + p, matches flat [B,P] layout
  const float c1 = K1c[bp];                 // revolutions per unit l
  const float c2 = K2c[bp];                 // revolutions per unit m
  const float ph = phi[bp];                 // phi already is a fraction of a turn

  const int tid   = threadIdx.x;
  const int mbase = (tid & 63) * 4;         // 64 threads cover one row of 256
  const int lrow  = tid >> 6;               // 4 rows in flight per iteration

  const float bm0 = c2 * (float)mbase;
  const float bm1 = bm0 + c2;
  const float bm2 = bm1 + c2;
  const float bm3 = bm2 + c2;

  float* dst = out + (size_t)bp * (S_ * S_) + lrow * S_ + mbase;
#pragma unroll 4
  for (int it = 0; it < S_ / 4; ++it) {
    const int l = it * 4 + lrow;
    const float tl = fmaf(c1, (float)l, ph);
    v4f v;
    v.x = __builtin_amdgcn_sinf(__builtin_amdgcn_fractf(tl + bm0));
    v.y = __builtin_amdgcn_sinf(__builtin_amdgcn_fractf(tl + bm1));
    v.z = __builtin_amdgcn_sinf(__builtin_amdgcn_fractf(tl + bm2));
    v.w = __builtin_amdgcn_sinf(__builtin_amdgcn_fractf(tl + bm3));
    // streaming store: 1 GiB output, never re-read -> keep it out of L2
    __builtin_nontemporal_store(v, (v4f*)(dst + (size_t)it * 4 * S_));
  }
}

// ---------------------------------------------------------------------------
extern "C" void kernel_launch(void* const* d_in, const int* in_sizes, int n_in,
                              void* d_out, int out_size, void* d_ws, size_t ws_size,
                              hipStream_t stream) {
  const float* Zg  = (const float*)d_in[0];  // [B,G]
  const float* phi = (const float*)d_in[1];  // [B,P,1,1]
  const float* Wl  = (const float*)d_in[2];  // [G,H]
  const float* bl  = (const float*)d_in[3];  // [H]
  const float* W1  = (const float*)d_in[4];  // [H,P]
  const float* b1  = (const float*)d_in[5];  // [P]
  const float* W2  = (const float*)d_in[6];  // [H,P]
  const float* b2  = (const float*)d_in[7];  // [P]
  float* out = (float*)d_out;                // [B,P,S,S]

  float* X   = (float*)d_ws;                 // [64,512]   = 128 KB
  float* K1c = X + B_ * H_;                  // [64,64]    =  16 KB
  float* K2c = K1c + B_ * P_;                // [64,64]    =  16 KB

  mlp_hidden_kernel<<<(B_ / 16) * (H_ / 16), 32, 0, stream>>>(Zg, Wl, bl, X);
  mlp_heads_kernel<<<2 * 4 * 4, 32, 0, stream>>>(X, W1, b1, W2, b2, K1c, K2c);
  periodic_sin_kernel<<<B_ * P_, 256, 0, stream>>>(K1c, K2c, phi, out);
}